// TipAdapterModule_42107859370333
// MI455X (gfx1250) — compile-verified
//
#include <hip/hip_runtime.h>

typedef float v2f __attribute__((ext_vector_type(2)));
typedef float v8f __attribute__((ext_vector_type(8)));

#define ALPHA_F 1.0f
#define BETA_F 5.5f
#define VLM_SCALE 100.0f
#define DIM 512
#define BATCH 2048
#define NSUP 100000
#define NCLS 400
#define KSTEPS (DIM / 4) /* 128 wmma_f32_16x16x4 steps over K=512 */
#define NSPLIT 4
#define BM 32 /* B rows per block in the main kernel */

// ---------------------------------------------------------------------------
// Kernel 1: out[b][c] = 100 * dot(embeds[b], text[c]).  One wave per 16x16
// output tile; exact fp32 WMMA (V_WMMA_F32_16X16X4_F32).
// B operand (text^T, K-major) staged in LDS in k-pair layout so each step is
// one ds_load_b64; A operand (embeds, M-major row-major) is a global b64 load.
// ---------------------------------------------------------------------------
__global__ __launch_bounds__(32) void tip_text_logits_kernel(
    const float* __restrict__ embeds, const float* __restrict__ text,
    float* __restrict__ out) {
  // [256 k-pairs][16 cols * 2] = 8192 floats = 32KB
  __shared__ float Bs[(DIM / 2) * 32];

  const int lane = threadIdx.x;      // 0..31
  const int b0 = blockIdx.x * 16;    // B tile
  const int c0 = blockIdx.y * 16;    // class tile

  // Stage text^T: element (c,k) -> Bs[(k>>1)*32 + 2c + (k&1)]
  for (int idx = lane; idx < 16 * DIM; idx += 32) {
    const int c = idx >> 9;        // / 512
    const int k = idx & (DIM - 1);
    Bs[(k >> 1) * 32 + 2 * c + (k & 1)] = text[(c0 + c) * DIM + k];
  }
  __syncthreads();

  const int kb2 = lane >> 4;          // 0 (K=0,1) / 1 (K=2,3) half
  const int kbase = kb2 << 1;
  const int ccol = lane & 15;
  const float* aptr = embeds + (b0 + ccol) * DIM + kbase;

  v8f acc = {};
  for (int kk = 0; kk < KSTEPS; ++kk) {
    v2f a = *(const v2f*)(aptr + kk * 4);                    // A: M=lane&15
    v2f b = *(const v2f*)(&Bs[(kk * 2 + kb2) * 32 + 2 * ccol]); // B: col=ccol
    acc = __builtin_amdgcn_wmma_f32_16x16x4_f32(
        /*neg_a=*/false, a, /*neg_b=*/false, b,
        /*c_mod=*/(short)0, acc, /*reuse_a=*/false, /*reuse_b=*/false);
  }

  // D layout: VGPR i: lanes0-15 -> M=i, lanes16-31 -> M=8+i; col = lane&15
  const int rowoff = kb2 * 8;
#pragma unroll
  for (int i = 0; i < 8; ++i)
    out[(b0 + rowoff + i) * NCLS + c0 + ccol] = VLM_SCALE * acc[i];
}

// ---------------------------------------------------------------------------
// Kernel 2: the big GEMM + exp kernel + one-hot scatter.
// Block = 256 threads (8 waves). Block owns BM=32 rows of B and 1/NSPLIT of N.
// Waves tiled as 4 (n-subtiles) x 2 (b-subtiles) -> 64 support rows / iter.
// aff tile via chained V_WMMA_F32_16X16X4_F32 (exact fp32), then
// w = exp(beta*aff - beta) scattered into LDS acc by label (ds_add_f32),
// finally flushed with global f32 atomics on top of the text logits.
// ---------------------------------------------------------------------------
__global__ __launch_bounds__(256) void tip_cache_logits_kernel(
    const float* __restrict__ embeds, const float* __restrict__ keys,
    const int* __restrict__ labels, float* __restrict__ out) {
  __shared__ float Bs[(DIM / 2) * (BM * 2)];  // embeds^T, 16384 f = 64KB
  __shared__ float accS[BM * 401];            // padded stride 401, ~51KB

  const int tid = threadIdx.x;
  const int lane = tid & 31;
  const int wave = tid >> 5;  // 0..7
  const int b0 = blockIdx.x * BM;

  // Stage embeds^T in k-pair layout: (b,k) -> Bs[(k>>1)*64 + 2b + (k&1)]
  for (int idx = tid; idx < BM * DIM; idx += 256) {
    const int b = idx >> 9;
    const int k = idx & (DIM - 1);
    Bs[(k >> 1) * 64 + 2 * b + (k & 1)] = embeds[(b0 + b) * DIM + k];
  }
  for (int idx = tid; idx < BM * 401; idx += 256) accS[idx] = 0.0f;
  __syncthreads();

  const int nsub = wave >> 1;  // which 16-row n subtile (0..3)
  const int bsub = wave & 1;   // which 16-col b subtile (0..1)
  const int ccol = lane & 15;
  const int bcol = bsub * 16 + ccol;  // b column in 0..31
  const int kb2 = lane >> 4;          // A/B half-K select
  const int kbase = kb2 << 1;

  const int Nper = (NSUP + NSPLIT - 1) / NSPLIT;  // 25000
  const int nStart = blockIdx.y * Nper;
  const int nEnd = (nStart + Nper < NSUP) ? (nStart + Nper) : NSUP;

  for (int nt = nStart; nt < nEnd; nt += 64) {
    const int n0 = nt + nsub * 16;
    int nrow = n0 + ccol;                          // A row (support index)
    int nclamp = (nrow < NSUP - 1) ? nrow : (NSUP - 1);
    const float* aptr = keys + nclamp * DIM + kbase;

    v8f acc = {};
    for (int kk = 0; kk < KSTEPS; ++kk) {
      v2f a = *(const v2f*)(aptr + kk * 4);  // A: cache_keys rows (M=n)
      v2f b = *(const v2f*)(&Bs[(kk * 2 + kb2) * 64 + 2 * bcol]);  // B: embeds^T
      acc = __builtin_amdgcn_wmma_f32_16x16x4_f32(
          false, a, false, b, (short)0, acc, false, false);
    }

    // aff element i: n = n0 + (lane>=16 ? 8 : 0) + i, b column = bcol
    const int nb = n0 + kb2 * 8;
#pragma unroll
    for (int i = 0; i < 8; ++i) {
      const int n = nb + i;
      if (n < nEnd) {
        const int lab = labels[n];  // uniform across lanes 0-15 (broadcast)
        const float w = __expf(fmaf(BETA_F, acc[i], -BETA_F));
        atomicAdd(&accS[bcol * 401 + lab], w);  // ds_add_f32
      }
    }
  }
  __syncthreads();

  // Flush per-block class accumulator onto the text logits.
  for (int idx = tid; idx < BM * NCLS; idx += 256) {
    const int b = idx / NCLS;
    const int c = idx - b * NCLS;
    __hip_atomic_fetch_add(&out[(b0 + b) * NCLS + c],
                           ALPHA_F * accS[b * 401 + c], __ATOMIC_RELAXED,
                           __HIP_MEMORY_SCOPE_AGENT);
  }
}

extern "C" void kernel_launch(void* const* d_in, const int* in_sizes, int n_in,
                              void* d_out, int out_size, void* d_ws,
                              size_t ws_size, hipStream_t stream) {
  const float* embeds = (const float*)d_in[0];      // [2048, 512]
  const float* cache_keys = (const float*)d_in[1];  // [100000, 512]
  const float* text_embeds = (const float*)d_in[2]; // [400, 512]
  const int* labels = (const int*)d_in[3];          // [100000]
  float* out = (float*)d_out;                       // [2048, 400]

  // Kernel 1 fully overwrites d_out with text logits (init for atomics).
  dim3 g1(BATCH / 16, NCLS / 16);  // 128 x 25 tiles, one wave each
  tip_text_logits_kernel<<<g1, 32, 0, stream>>>(embeds, text_embeds, out);

  // Kernel 2: 64 B-tiles x 4 N-slices = 256 blocks (~1 per WGP).
  dim3 g2(BATCH / BM, NSPLIT);
  tip_cache_logits_kernel<<<g2, 256, 0, stream>>>(embeds, cache_keys, labels,
                                                  out);
}